// HausdorffDTLoss_2576980377823
// MI455X (gfx1250) — compile-verified
//
#include <hip/hip_runtime.h>

// Problem constants (B=4, C=1, H=W=320)
#define HW    320
#define BB    4
#define NPIX  (HW * HW)            // 102400 per image
#define G2_FLOATS (4 * BB * NPIX)  // 4 fields x B x H x W = 1,638,400 floats
#define NBLK3 400                  // column-pass blocks (B=4 x 10 col-tiles x 10 row-tiles)

typedef float v2f __attribute__((ext_vector_type(2)));
typedef float v8f __attribute__((ext_vector_type(8)));

// ---------------------------------------------------------------------------
// Kernel 1: has_fg flag per image (8 images: pred b0..3, target b0..3)
// ---------------------------------------------------------------------------
__global__ void __launch_bounds__(256)
hasfg_kernel(const float* __restrict__ pred, const float* __restrict__ tgt,
             int* __restrict__ flags) {
  const int img = blockIdx.x;  // 0..7
  const float* src = ((img < 4) ? pred : tgt) + (size_t)(img & 3) * NPIX;
  int any = 0;
  for (int i = threadIdx.x; i < NPIX; i += 256)
    any |= (src[i] > 0.5f) ? 1 : 0;
  __shared__ int s[256];
  s[threadIdx.x] = any;
  __syncthreads();
  for (int st = 128; st > 0; st >>= 1) {
    if (threadIdx.x < st) s[threadIdx.x] |= s[threadIdx.x + st];
    __syncthreads();
  }
  if (threadIdx.x == 0) flags[img] = s[0];
}

// ---------------------------------------------------------------------------
// Kernel 2: row pass. g = 1D distance along row to nearest "True" of z,
// exactly mirroring the reference cummax/cummin with +-BIG sentinels.
// One 320-thread block per (field, b, row); stores g^2.
// fields: 0=pred fg(z=!mask), 1=pred bg(z=mask), 2=tgt fg, 3=tgt bg
// ---------------------------------------------------------------------------
__global__ void __launch_bounds__(HW)
rowpass_kernel(const float* __restrict__ pred, const float* __restrict__ tgt,
               float* __restrict__ g2) {
  const int fb  = blockIdx.x;
  const int f   = fb / (BB * HW);
  const int rem = fb - f * (BB * HW);
  const int b   = rem / HW;
  const int row = rem - b * HW;
  const int tid = threadIdx.x;  // column 0..319

  const float* img = (((f >> 1) == 0) ? pred : tgt) + (size_t)b * NPIX + (size_t)row * HW;
  const bool mask = img[tid] > 0.5f;
  const bool z    = (f & 1) ? mask : !mask;

  __shared__ float s[HW];
  const float fi = (float)tid;

  // inclusive max-scan of (z ? i : -1e6)
  float v = z ? fi : -1.0e6f;
  s[tid] = v;
  __syncthreads();
#pragma unroll
  for (int off = 1; off < HW; off <<= 1) {
    float o = (tid >= off) ? s[tid - off] : -1.0e30f;
    __syncthreads();
    v = fmaxf(v, o);
    s[tid] = v;
    __syncthreads();
  }
  const float left = v;

  // reverse inclusive min-scan of (z ? i : 2e6)
  float w = z ? fi : 2.0e6f;
  s[tid] = w;
  __syncthreads();
#pragma unroll
  for (int off = 1; off < HW; off <<= 1) {
    float o = (tid + off < HW) ? s[tid + off] : 3.0e30f;
    __syncthreads();
    w = fminf(w, o);
    s[tid] = w;
    __syncthreads();
  }
  const float right = w;

  const float g = fminf(fi - left, right - fi);
  g2[(size_t)((f * BB + b) * HW + row) * HW + tid] = g * g;
}

// ---------------------------------------------------------------------------
// Kernel 3: column pass d2[i,j] = min_k g2[k,j] + (i-k)^2 for the 4 fields,
// fused with sqrt/field/err*dist and a deterministic block reduction.
// Block tile: 32 cols x 32 rows; 256 threads (jl = tid&31, isub = tid>>5,
// each thread owns rows i0+isub+8r, r=0..3). LDS tile: 320x32 f32 = 40 KB.
// ---------------------------------------------------------------------------
__global__ void __launch_bounds__(256)
colpass_kernel(const float* __restrict__ pred, const float* __restrict__ tgt,
               const float* __restrict__ g2, const int* __restrict__ flags,
               float* __restrict__ partials) {
  const int blk = blockIdx.x;       // 0..399
  const int b   = blk / 100;
  const int r2  = blk - b * 100;
  const int jt  = r2 / 10;
  const int it  = r2 - jt * 10;
  const int j0  = jt * 32;
  const int i0  = it * 32;
  const int tid = threadIdx.x;
  const int jl  = tid & 31;
  const int isub = tid >> 5;        // 0..7

  __shared__ float tile[HW * 32];   // 40 KB

  float acc[4][4];
#pragma unroll
  for (int f = 0; f < 4; ++f)
#pragma unroll
    for (int r = 0; r < 4; ++r) acc[f][r] = 3.0e38f;

  for (int f = 0; f < 4; ++f) {
    const float* gf = g2 + (size_t)(f * BB + b) * NPIX;
    __syncthreads();
    for (int idx = tid; idx < HW * 32; idx += 256) {
      const int k = idx >> 5, jj = idx & 31;
      tile[idx] = gf[(size_t)k * HW + j0 + jj];   // coalesced 128B row segments
    }
    __syncthreads();
    const float fi0 = (float)(i0 + isub);
    for (int k = 0; k < HW; ++k) {
      const float gv = tile[k * 32 + jl];
      const float fk = (float)k;
#pragma unroll
      for (int r = 0; r < 4; ++r) {
        const float diff = (fi0 + (float)(8 * r)) - fk;          // exact
        acc[f][r] = fminf(acc[f][r], __builtin_fmaf(diff, diff, gv));
      }
    }
  }

  const int hp = flags[b];
  const int ht = flags[4 + b];
  float lsum = 0.0f;
#pragma unroll
  for (int r = 0; r < 4; ++r) {
    const int i = i0 + isub + 8 * r;
    const float fp = hp ? (__fsqrt_rn(acc[0][r]) + __fsqrt_rn(acc[1][r])) : 0.0f;
    const float ft = ht ? (__fsqrt_rn(acc[2][r]) + __fsqrt_rn(acc[3][r])) : 0.0f;
    const float dist = fp * fp + ft * ft;              // ALPHA = 2
    const size_t off = (size_t)b * NPIX + (size_t)i * HW + j0 + jl;
    const float pe = pred[off] - tgt[off];
    lsum += (pe * pe) * dist;
  }

  __syncthreads();
  tile[tid] = lsum;
  __syncthreads();
  for (int st = 128; st > 0; st >>= 1) {
    if (tid < st) tile[tid] += tile[tid + st];
    __syncthreads();
  }
  if (tid == 0) partials[blk] = tile[0];
}

// ---------------------------------------------------------------------------
// Kernel 4: deterministic final reduction of the 400 block partials using
// V_WMMA_F32_16X16X4_F32 with B = ones (row-sum engine, exact fp32 adds).
// One wave (32 lanes), EXEC all ones, no divergence around the WMMA.
// A layout (16x4 f32): lane<16 -> (M=lane,K=0/1) in v0/v1; lane>=16 -> K=2/3.
// D[m,n] = rowsum_m (replicated over n); total = sum over M of (N=0) column:
// lane0 holds M=r in VGPR r (r=0..7), lane16 holds M=8+r.
// ---------------------------------------------------------------------------
__global__ void __launch_bounds__(32)
final_kernel(const float* __restrict__ partials, float* __restrict__ out) {
  const int l = threadIdx.x;  // 0..31
  v2f bmat;
  bmat.x = 1.0f;
  bmat.y = 1.0f;
  v8f c = {};
  const int iters = (NBLK3 + 63) / 64;  // 7
  for (int itn = 0; itn < iters; ++itn) {
    const int base = itn * 64;
    const int i0 = base + l;
    const int i1 = base + 32 + l;
    v2f a;
    a.x = (i0 < NBLK3) ? partials[i0] : 0.0f;
    a.y = (i1 < NBLK3) ? partials[i1] : 0.0f;
    c = __builtin_amdgcn_wmma_f32_16x16x4_f32(false, a, false, bmat,
                                              (short)0, c, false, false);
  }
  __shared__ float s[32 * 8];
#pragma unroll
  for (int r = 0; r < 8; ++r) s[l * 8 + r] = c[r];
  __syncthreads();
  if (l == 0) {
    float tot = 0.0f;
#pragma unroll
    for (int r = 0; r < 8; ++r) tot += s[0 * 8 + r] + s[16 * 8 + r];
    out[0] = tot / 409600.0f;  // mean over B*C*H*W
  }
}

// ---------------------------------------------------------------------------
extern "C" void kernel_launch(void* const* d_in, const int* in_sizes, int n_in,
                              void* d_out, int out_size, void* d_ws, size_t ws_size,
                              hipStream_t stream) {
  (void)in_sizes; (void)n_in; (void)out_size; (void)ws_size;
  const float* pred = (const float*)d_in[0];
  const float* tgt  = (const float*)d_in[1];

  // workspace layout: [g2: 6,553,600 B][flags: 32 B][partials: 1600 B]
  float* g2       = (float*)d_ws;
  int*   flags    = (int*)((char*)d_ws + (size_t)G2_FLOATS * 4);
  float* partials = (float*)((char*)d_ws + (size_t)G2_FLOATS * 4 + 32);

  hasfg_kernel<<<8, 256, 0, stream>>>(pred, tgt, flags);
  rowpass_kernel<<<4 * BB * HW, HW, 0, stream>>>(pred, tgt, g2);
  colpass_kernel<<<NBLK3, 256, 0, stream>>>(pred, tgt, g2, flags, partials);
  final_kernel<<<1, 32, 0, stream>>>(partials, (float*)d_out);
}